// Net_70600672411795
// MI455X (gfx1250) — compile-verified
//
#include <hip/hip_runtime.h>
#include <hip/hip_bf16.h>
#include <math.h>
#include <stdint.h>

// ---------------- problem constants (match reference) ----------------
#define N_NODES 50000
#define N_EDGES 1600000
#define E_TOT   (N_EDGES + N_NODES)   // edges + self loops
#define F_IN    2304
#define H1      8
#define D1      8
#define F_MID   64                    // H1*D1
#define C_OUT   5

typedef float v2f __attribute__((ext_vector_type(2)));
typedef float v8f __attribute__((ext_vector_type(8)));

static inline int cdiv(int a, int b) { return (a + b - 1) / b; }

// ---------------- helpers ----------------
__device__ __forceinline__ void edge_sd(const int* __restrict__ ei, int e, int& s, int& d) {
    if (e < N_EDGES) { s = ei[e]; d = ei[N_EDGES + e]; }
    else             { s = e - N_EDGES; d = s; }      // self loop
}

// float atomic max via signed-max / unsigned-min trick (init to -inf)
__device__ __forceinline__ void atomicMaxFloat(float* addr, float val) {
    if (val >= 0.0f) atomicMax((int*)addr, __float_as_int(val));
    else             atomicMin((unsigned int*)addr, __float_as_uint(val));
}

__device__ __forceinline__ float lrelu02(float v) { return v > 0.0f ? v : 0.2f * v; }

// async global->LDS copy of 16 bytes (CDNA5 VGLOBAL async op, ASYNCcnt-tracked)
__device__ __forceinline__ void async_g2l_b128(unsigned lds_off, const void* gptr) {
    unsigned long long ga = (unsigned long long)(uintptr_t)gptr;
    asm volatile("global_load_async_to_lds_b128 %0, %1, off"
                 :: "v"(lds_off), "v"(ga) : "memory");
}
__device__ __forceinline__ void wait_async0() {
    asm volatile("s_wait_asynccnt 0x0" ::: "memory");
}
__device__ __forceinline__ unsigned lds_off_of(const void* p) {
    // low 32 bits of generic pointer to LDS = LDS byte offset
    return (unsigned)(uintptr_t)p;
}

// ---------------- init ----------------
__global__ void fill_kernel(float* __restrict__ p, int n, float v) {
    int i = blockIdx.x * blockDim.x + threadIdx.x;
    if (i < n) p[i] = v;
}

// ---------------- layer 1 GEMM: h1[N,64] = x[N,2304] @ W1[2304,64] ----------------
// Block = 128 threads = 4 waves; each wave owns a 16-row tile, all 64 cols
// (4 fp32 WMMA accumulators). K is chunked (KC=64) and double-buffered in LDS:
//   Wbuf: shared W1 chunk (row stride 80 dwords -> conflict-free B reads, 16B-aligned)
//   Xbuf: per-wave 16xKC x-tile (row stride 68 dwords -> conflict-free A reads)
// Chunks are fetched with global_load_async_to_lds_b128 and overlap WMMA compute.
__global__ __launch_bounds__(128)
void gemm1_wmma(const float* __restrict__ x, const float* __restrict__ W,
                float* __restrict__ h) {
    constexpr int KC  = 64;
    constexpr int NCH = F_IN / KC;       // 36
    constexpr int WS  = 80;              // Wbuf row stride (dwords)
    constexpr int XS  = 68;              // Xbuf row stride (dwords)

    __shared__ float Wbuf[2][KC * WS];   // 2 * 20.0 KB
    __shared__ float Xbuf[2][4][16 * XS];// 2 * 17.0 KB

    const int tid   = threadIdx.x;
    const int lane  = tid & 31;
    const int wid   = tid >> 5;
    const int tile  = blockIdx.x * 4 + wid;
    const int row0  = tile * 16;
    const bool valid = row0 < N_NODES;   // wave-uniform
    const int m     = lane & 15;
    const int khalf = (lane >> 4) << 1;  // 0 or 2

    v8f acc0 = {}, acc1 = {}, acc2 = {}, acc3 = {};

    auto issue_chunk = [&](int kc, int buf) {
        // W1 chunk: 64x64 floats, cooperative across 128 threads, 8x b128 each
        const float* gw = W + (size_t)kc * F_MID;
#pragma unroll
        for (int i = 0; i < 8; ++i) {
            int u = tid + i * 128;               // 0..1023 (b128 units)
            int r = u >> 4;                      // 0..63
            int c = (u & 15) << 2;               // 0..60
            async_g2l_b128(lds_off_of(&Wbuf[buf][r * WS + c]), gw + r * F_MID + c);
        }
        if (valid) {
            // x tile: 16x64 floats for this wave, 8x b128 per lane
            const float* gx = x + (size_t)row0 * F_IN + kc;
#pragma unroll
            for (int i = 0; i < 8; ++i) {
                int u = lane + i * 32;           // 0..255
                int r = u >> 4;                  // 0..15
                int c = (u & 15) << 2;           // 0..60
                async_g2l_b128(lds_off_of(&Xbuf[buf][wid][r * XS + c]),
                               gx + (size_t)r * F_IN + c);
            }
        }
    };

    issue_chunk(0, 0);
    for (int ch = 0; ch < NCH; ++ch) {
        wait_async0();                   // this wave's chunk `ch` data has landed
        __syncthreads();                 // all waves' writes visible, prior reads done
        if (ch + 1 < NCH) issue_chunk((ch + 1) * KC, (ch + 1) & 1);  // overlap copy
        const int buf = ch & 1;
        if (valid) {
            const float* Xw = &Xbuf[buf][wid][0];
            const float* Wb = &Wbuf[buf][0];
#pragma unroll
            for (int kk = 0; kk < KC; kk += 4) {
                v2f a = *(const v2f*)(Xw + m * XS + kk + khalf);
                const float* wk = Wb + (kk + khalf) * WS + m;
                v2f b0 = { wk[0],  wk[WS + 0]  };
                v2f b1 = { wk[16], wk[WS + 16] };
                v2f b2 = { wk[32], wk[WS + 32] };
                v2f b3 = { wk[48], wk[WS + 48] };
                acc0 = __builtin_amdgcn_wmma_f32_16x16x4_f32(false, a, false, b0, (short)0, acc0, false, false);
                acc1 = __builtin_amdgcn_wmma_f32_16x16x4_f32(false, a, false, b1, (short)0, acc1, false, false);
                acc2 = __builtin_amdgcn_wmma_f32_16x16x4_f32(false, a, false, b2, (short)0, acc2, false, false);
                acc3 = __builtin_amdgcn_wmma_f32_16x16x4_f32(false, a, false, b3, (short)0, acc3, false, false);
            }
        }
    }

    if (valid) {
        const int mbase = (lane >> 4) << 3;  // 0 or 8
#pragma unroll
        for (int v = 0; v < 8; ++v) {
            float* hrow = h + (size_t)(row0 + mbase + v) * F_MID + m;
            hrow[0]  = acc0[v];
            hrow[16] = acc1[v];
            hrow[32] = acc2[v];
            hrow[48] = acc3[v];
        }
    }
}

// ---------------- per-node attention coefficients, layer 1 ----------------
__global__ void att_coef1(const float* __restrict__ h1,
                          const float* __restrict__ att_src, const float* __restrict__ att_dst,
                          float* __restrict__ a_s, float* __restrict__ a_d) {
    int i = blockIdx.x * blockDim.x + threadIdx.x;    // over N*H1
    if (i >= N_NODES * H1) return;
    int n = i >> 3, hh = i & 7;
    const float* hv = h1 + (size_t)n * F_MID + hh * D1;
    const float* ws = att_src + hh * D1;
    const float* wd = att_dst + hh * D1;
    float s = 0.0f, d = 0.0f;
#pragma unroll
    for (int j = 0; j < D1; ++j) { float v = hv[j]; s += v * ws[j]; d += v * wd[j]; }
    a_s[i] = s; a_d[i] = d;
}

// ---------------- layer-1 edge passes (H1 heads) ----------------
__global__ void edge_max1(const int* __restrict__ ei, const float* __restrict__ a_s,
                          const float* __restrict__ a_d, float* __restrict__ m1) {
    int i = blockIdx.x * blockDim.x + threadIdx.x;    // over E_TOT*H1
    if (i >= E_TOT * H1) return;
    int e = i >> 3, hh = i & 7, s, d;
    edge_sd(ei, e, s, d);
    float l = lrelu02(a_s[s * H1 + hh] + a_d[d * H1 + hh]);
    atomicMaxFloat(&m1[d * H1 + hh], l);
}

__global__ void edge_sum1(const int* __restrict__ ei, const float* __restrict__ a_s,
                          const float* __restrict__ a_d, const float* __restrict__ m1,
                          float* __restrict__ z1) {
    int i = blockIdx.x * blockDim.x + threadIdx.x;
    if (i >= E_TOT * H1) return;
    int e = i >> 3, hh = i & 7, s, d;
    edge_sd(ei, e, s, d);
    float l = lrelu02(a_s[s * H1 + hh] + a_d[d * H1 + hh]);
    atomicAdd(&z1[d * H1 + hh], expf(l - m1[d * H1 + hh]));
}

__global__ void edge_scatter1(const int* __restrict__ ei, const float* __restrict__ a_s,
                              const float* __restrict__ a_d, const float* __restrict__ m1,
                              const float* __restrict__ z1, const float* __restrict__ h1,
                              float* __restrict__ out) {
    int i = blockIdx.x * blockDim.x + threadIdx.x;    // over E_TOT*64 (~106M, fits int)
    if (i >= E_TOT * F_MID) return;
    int e = i >> 6, j = i & 63, hh = j >> 3, s, d;
    edge_sd(ei, e, s, d);
    float l = lrelu02(a_s[s * H1 + hh] + a_d[d * H1 + hh]);
    float alpha = expf(l - m1[d * H1 + hh]) / (z1[d * H1 + hh] + 1e-16f);
    atomicAdd(&out[(size_t)d * F_MID + j], h1[(size_t)s * F_MID + j] * alpha);
}

__global__ void add_bias1(float* __restrict__ attn1, const float* __restrict__ b1) {
    int i = blockIdx.x * blockDim.x + threadIdx.x;
    if (i >= N_NODES * F_MID) return;
    attn1[i] += b1[i & 63];
}

// ---------------- layer 2 GEMM: h2[N,5(pad8)] = attn1[N,64] @ W2[64,5] ----------------
__global__ __launch_bounds__(128)
void gemm2_wmma(const float* __restrict__ hin, const float* __restrict__ W2,
                float* __restrict__ h2) {
    const int lane  = threadIdx.x & 31;
    const int tile  = blockIdx.x * 4 + (threadIdx.x >> 5);
    if (tile * 16 >= N_NODES) return;
    const int row0  = tile * 16;
    const int m     = lane & 15;
    const int khalf = (lane >> 4) << 1;

    const float* __restrict__ arow = hin + (size_t)(row0 + m) * F_MID + khalf;
    v8f acc = {};
    for (int k = 0; k < F_MID; k += 4) {
        v2f a = *(const v2f*)(arow + k);
        const float* wk = W2 + (size_t)(k + khalf) * C_OUT;
        v2f b;
        b.x = (m < C_OUT) ? wk[m] : 0.0f;
        b.y = (m < C_OUT) ? wk[C_OUT + m] : 0.0f;
        acc = __builtin_amdgcn_wmma_f32_16x16x4_f32(false, a, false, b, (short)0, acc, false, false);
    }
    if (m < C_OUT) {
        const int mbase = (lane >> 4) << 3;
#pragma unroll
        for (int v = 0; v < 8; ++v)
            h2[(size_t)(row0 + mbase + v) * 8 + m] = acc[v];
    }
}

// ---------------- per-node attention coefficients, layer 2 (1 head, C=5) ----------------
__global__ void att_coef2(const float* __restrict__ h2,
                          const float* __restrict__ att_src, const float* __restrict__ att_dst,
                          float* __restrict__ a_s, float* __restrict__ a_d) {
    int n = blockIdx.x * blockDim.x + threadIdx.x;
    if (n >= N_NODES) return;
    float s = 0.0f, d = 0.0f;
#pragma unroll
    for (int c = 0; c < C_OUT; ++c) { float v = h2[(size_t)n * 8 + c]; s += v * att_src[c]; d += v * att_dst[c]; }
    a_s[n] = s; a_d[n] = d;
}

__global__ void edge_max2(const int* __restrict__ ei, const float* __restrict__ a_s,
                          const float* __restrict__ a_d, float* __restrict__ m2) {
    int e = blockIdx.x * blockDim.x + threadIdx.x;
    if (e >= E_TOT) return;
    int s, d; edge_sd(ei, e, s, d);
    atomicMaxFloat(&m2[d], lrelu02(a_s[s] + a_d[d]));
}

__global__ void edge_sum2(const int* __restrict__ ei, const float* __restrict__ a_s,
                          const float* __restrict__ a_d, const float* __restrict__ m2,
                          float* __restrict__ z2) {
    int e = blockIdx.x * blockDim.x + threadIdx.x;
    if (e >= E_TOT) return;
    int s, d; edge_sd(ei, e, s, d);
    float l = lrelu02(a_s[s] + a_d[d]);
    atomicAdd(&z2[d], expf(l - m2[d]));
}

__global__ void edge_scatter2(const int* __restrict__ ei, const float* __restrict__ a_s,
                              const float* __restrict__ a_d, const float* __restrict__ m2,
                              const float* __restrict__ z2, const float* __restrict__ h2,
                              float* __restrict__ out) {
    int i = blockIdx.x * blockDim.x + threadIdx.x;    // over E_TOT*8, lanes c<5 active
    if (i >= E_TOT * 8) return;
    int e = i >> 3, c = i & 7;
    if (c >= C_OUT) return;
    int s, d; edge_sd(ei, e, s, d);
    float l = lrelu02(a_s[s] + a_d[d]);
    float alpha = expf(l - m2[d]) / (z2[d] + 1e-16f);
    atomicAdd(&out[(size_t)d * 8 + c], h2[(size_t)s * 8 + c] * alpha);
}

// ---------------- final: bias + log_softmax over 5 classes ----------------
__global__ void final_logsoftmax(const float* __restrict__ attn2, const float* __restrict__ b2,
                                 float* __restrict__ out) {
    int n = blockIdx.x * blockDim.x + threadIdx.x;
    if (n >= N_NODES) return;
    float v[C_OUT], mx = -INFINITY;
#pragma unroll
    for (int c = 0; c < C_OUT; ++c) { v[c] = attn2[(size_t)n * 8 + c] + b2[c]; mx = fmaxf(mx, v[c]); }
    float sum = 0.0f;
#pragma unroll
    for (int c = 0; c < C_OUT; ++c) sum += expf(v[c] - mx);
    float lse = mx + logf(sum);
#pragma unroll
    for (int c = 0; c < C_OUT; ++c) out[n * C_OUT + c] = v[c] - lse;
}

// ---------------- launcher ----------------
extern "C" void kernel_launch(void* const* d_in, const int* in_sizes, int n_in,
                              void* d_out, int out_size, void* d_ws, size_t ws_size,
                              hipStream_t stream) {
    const float* x   = (const float*)d_in[0];
    const int*   ei  = (const int*)  d_in[1];
    const float* W1  = (const float*)d_in[2];
    const float* as1 = (const float*)d_in[3];
    const float* ad1 = (const float*)d_in[4];
    const float* b1  = (const float*)d_in[5];
    const float* W2  = (const float*)d_in[6];
    const float* as2 = (const float*)d_in[7];
    const float* ad2 = (const float*)d_in[8];
    const float* b2  = (const float*)d_in[9];
    float* out = (float*)d_out;

    // workspace carve-up (~36 MB of fp32)
    float* ws    = (float*)d_ws;
    float* h1    = ws; ws += (size_t)N_NODES * F_MID;
    float* a_s1  = ws; ws += (size_t)N_NODES * H1;
    float* a_d1  = ws; ws += (size_t)N_NODES * H1;
    float* m1    = ws; ws += (size_t)N_NODES * H1;
    float* z1    = ws; ws += (size_t)N_NODES * H1;
    float* attn1 = ws; ws += (size_t)N_NODES * F_MID;
    float* h2    = ws; ws += (size_t)N_NODES * 8;
    float* a_s2  = ws; ws += (size_t)N_NODES;
    float* a_d2  = ws; ws += (size_t)N_NODES;
    float* m2    = ws; ws += (size_t)N_NODES;
    float* z2    = ws; ws += (size_t)N_NODES;
    float* attn2 = ws; ws += (size_t)N_NODES * 8;

    const int T = 256;

    // -------- init accumulators (fresh every call) --------
    fill_kernel<<<cdiv(N_NODES * H1, T), T, 0, stream>>>(m1, N_NODES * H1, -INFINITY);
    fill_kernel<<<cdiv(N_NODES * H1, T), T, 0, stream>>>(z1, N_NODES * H1, 0.0f);
    fill_kernel<<<cdiv(N_NODES * F_MID, T), T, 0, stream>>>(attn1, N_NODES * F_MID, 0.0f);
    fill_kernel<<<cdiv(N_NODES, T), T, 0, stream>>>(m2, N_NODES, -INFINITY);
    fill_kernel<<<cdiv(N_NODES, T), T, 0, stream>>>(z2, N_NODES, 0.0f);
    fill_kernel<<<cdiv(N_NODES * 8, T), T, 0, stream>>>(attn2, N_NODES * 8, 0.0f);

    // -------- layer 1 --------
    gemm1_wmma<<<cdiv(N_NODES, 64), 128, 0, stream>>>(x, W1, h1);
    att_coef1<<<cdiv(N_NODES * H1, T), T, 0, stream>>>(h1, as1, ad1, a_s1, a_d1);
    edge_max1<<<cdiv(E_TOT * H1, T), T, 0, stream>>>(ei, a_s1, a_d1, m1);
    edge_sum1<<<cdiv(E_TOT * H1, T), T, 0, stream>>>(ei, a_s1, a_d1, m1, z1);
    edge_scatter1<<<cdiv(E_TOT * F_MID, T), T, 0, stream>>>(ei, a_s1, a_d1, m1, z1, h1, attn1);
    add_bias1<<<cdiv(N_NODES * F_MID, T), T, 0, stream>>>(attn1, b1);

    // -------- layer 2 --------
    gemm2_wmma<<<cdiv(N_NODES, 64), 128, 0, stream>>>(attn1, W2, h2);
    att_coef2<<<cdiv(N_NODES, T), T, 0, stream>>>(h2, as2, ad2, a_s2, a_d2);
    edge_max2<<<cdiv(E_TOT, T), T, 0, stream>>>(ei, a_s2, a_d2, m2);
    edge_sum2<<<cdiv(E_TOT, T), T, 0, stream>>>(ei, a_s2, a_d2, m2, z2);
    edge_scatter2<<<cdiv(E_TOT * 8, T), T, 0, stream>>>(ei, a_s2, a_d2, m2, z2, h2, attn2);

    // -------- output --------
    final_logsoftmax<<<cdiv(N_NODES, T), T, 0, stream>>>(attn2, b2, out);
}